// Model_60713657696964
// MI455X (gfx1250) — compile-verified
//
#include <hip/hip_runtime.h>

// Streaming AMP grad-unscale + non-finite check for MI455X (gfx1250, wave32).
//
// Traffic: 127.40M fp32 in + 127.40M fp32 out ~= 1.02 GB  ->  ~44 us @ 23.3 TB/s.
// Memory-bound by >1e5x over compute. Strategy:
//   - block-uniform segment resolution (scalar branch, SGPR compares)
//   - fast path: 8 NT b128 loads (clause) + 8 NT b128 stores, no per-access
//     select / bounds checks (reference tile boundaries are tile-aligned)
//   - slow generic path only for straddling/partial blocks (never taken here)
//   - found_inf: rare-event conditional LDS atomicOr + idempotent 1.0f store

typedef __attribute__((ext_vector_type(4))) float    v4f;
typedef __attribute__((ext_vector_type(4))) unsigned v4u;

#define VPT 8u  // vec4s per thread (128 B per lane)

__device__ __forceinline__ bool any_nonfinite4(v4f v) {
    const v4u b = __builtin_bit_cast(v4u, v);
    bool nf = false;
    #pragma unroll
    for (int c = 0; c < 4; ++c) {
        // inf|nan <=> exponent field all ones.
        nf |= ((b[c] & 0x7F800000u) == 0x7F800000u);
    }
    return nf;
}

__global__ void amp_init_kernel(float* __restrict__ found_inf) {
    // Single thread: reset found_inf before the main pass (d_out is poisoned).
    *found_inf = 0.0f;
}

__global__ void __launch_bounds__(256)
amp_unscale_kernel(const float* __restrict__ g0,
                   const float* __restrict__ g1,
                   const float* __restrict__ g2,
                   const float* __restrict__ scale_ptr,
                   float* __restrict__ out,
                   float* __restrict__ found_inf,
                   unsigned nv0,      // g0 length in float4 units
                   unsigned nv01,     // g0+g1 length in float4 units
                   unsigned nv)       // total length in float4 units
{
    __shared__ int s_flag;
    if (threadIdx.x == 0) s_flag = 0;
    __syncthreads();

    const unsigned per_block = blockDim.x * VPT;
    const unsigned tile      = blockIdx.x * per_block;   // first vec4 of block
    const unsigned tile_end  = tile + per_block;         // exclusive

    // Uniform address -> scalarized load; broadcast to all 32 lanes.
    const float s  = *scale_ptr;
    const bool  sz = (s == 0.0f);

    bool nonfinite = false;

    // Block-uniform segment resolution (SGPR compares -> scalar branch).
    const v4f* src = nullptr;
    if (tile_end <= nv0)                          src = (const v4f*)g0 + tile;
    else if (tile >= nv0  && tile_end <= nv01)    src = (const v4f*)g1 + (tile - nv0);
    else if (tile >= nv01 && tile_end <= nv)      src = (const v4f*)g2 + (tile - nv01);

    if (src) {
        // ---- fast path: whole tile in one tensor, fully in range ----
        v4f vals[VPT];
        #pragma unroll
        for (unsigned k = 0; k < VPT; ++k) {
            // NT load (th:TH_LOAD_NT); 8 loads back-to-back -> VMEM clause.
            vals[k] = __builtin_nontemporal_load(src + threadIdx.x + k * blockDim.x);
        }
        #pragma unroll
        for (unsigned k = 0; k < VPT; ++k) {
            nonfinite |= any_nonfinite4(vals[k]);
            const v4f r = sz ? (v4f)(0.0f) : (vals[k] * s);
            // NT store (th:TH_STORE_NT); output never re-read.
            __builtin_nontemporal_store(
                r, (v4f*)out + tile + threadIdx.x + k * blockDim.x);
        }
    } else {
        // ---- generic path: tile straddles a boundary or the end ----
        #pragma unroll
        for (unsigned k = 0; k < VPT; ++k) {
            const unsigned t = tile + threadIdx.x + k * blockDim.x;
            if (t < nv) {
                const float* base = g0;
                unsigned     off  = t;
                if (t >= nv0)  { base = g1; off = t - nv0;  }
                if (t >= nv01) { base = g2; off = t - nv01; }
                const v4f v = __builtin_nontemporal_load((const v4f*)base + off);
                nonfinite |= any_nonfinite4(v);
                const v4f r = sz ? (v4f)(0.0f) : (v * s);
                __builtin_nontemporal_store(r, (v4f*)out + t);
            }
        }
    }

    // Rare-event block reduction: DS atomic only fires when a lane saw a
    // non-finite value (zero DS traffic on clean data).
    if (nonfinite) atomicOr(&s_flag, 1);
    __syncthreads();

    // All writers store the identical value 1.0f -> race-free without atomics.
    if (threadIdx.x == 0 && s_flag) *found_inf = 1.0f;
}

// Scalar tail for totals not divisible by 4 (empty for the reference shapes).
__global__ void amp_tail_kernel(const float* __restrict__ g2,
                                const float* __restrict__ scale_ptr,
                                float* __restrict__ out,
                                float* __restrict__ found_inf,
                                unsigned n01,      // elements in g0+g1
                                unsigned tail_lo,  // first leftover flat index
                                unsigned total)    // total elements
{
    const float s  = *scale_ptr;
    const bool  sz = (s == 0.0f);
    bool nonfinite = false;
    for (unsigned i = tail_lo + threadIdx.x; i < total; i += blockDim.x) {
        const float v = g2[i - n01];  // tail always lands in the last tensor
        const unsigned b = __builtin_bit_cast(unsigned, v);
        nonfinite |= ((b & 0x7F800000u) == 0x7F800000u);
        out[i] = sz ? 0.0f : v * s;
    }
    if (nonfinite) *found_inf = 1.0f;
}

extern "C" void kernel_launch(void* const* d_in, const int* in_sizes, int n_in,
                              void* d_out, int out_size, void* d_ws, size_t ws_size,
                              hipStream_t stream) {
    (void)n_in; (void)d_ws; (void)ws_size; (void)out_size;

    const float* g0    = (const float*)d_in[0];
    const float* g1    = (const float*)d_in[1];
    const float* g2    = (const float*)d_in[2];
    // d_in[3] = found_inf input tensor: ignored (reference recomputes it).
    const float* scale = (const float*)d_in[4];

    const unsigned long long n0 = (unsigned long long)in_sizes[0];
    const unsigned long long n1 = (unsigned long long)in_sizes[1];
    const unsigned long long n2 = (unsigned long long)in_sizes[2];
    const unsigned long long total = n0 + n1 + n2;   // 127,401,984

    float* out       = (float*)d_out;
    float* found_inf = out + total;                  // last output element

    const unsigned nv0  = (unsigned)(n0 / 4);
    const unsigned nv01 = (unsigned)((n0 + n1) / 4);
    const unsigned nv   = (unsigned)(total / 4);     // all shapes %4 == 0

    // 1) zero found_inf (stream order serializes vs. the main pass)
    amp_init_kernel<<<1, 1, 0, stream>>>(found_inf);

    // 2) main streaming pass: VPT float4s per thread
    const unsigned threads    = 256;
    const unsigned per_block  = threads * VPT;       // 2048 vec4s / block
    const unsigned blocks     = (nv + per_block - 1) / per_block;
    amp_unscale_kernel<<<blocks, threads, 0, stream>>>(
        g0, g1, g2, scale, out, found_inf, nv0, nv01, nv);

    // 3) scalar tail (no-op for the reference shapes, kept for robustness)
    const unsigned long long tail_lo = (unsigned long long)nv * 4ull;
    if (tail_lo < total) {
        amp_tail_kernel<<<1, 64, 0, stream>>>(
            g2, scale, out, found_inf,
            (unsigned)(n0 + n1), (unsigned)tail_lo, (unsigned)total);
    }
}